// BSplineEncoder_20418274525442
// MI455X (gfx1250) — compile-verified
//
#include <hip/hip_runtime.h>

typedef float v2f __attribute__((ext_vector_type(2)));
typedef float v4f __attribute__((ext_vector_type(4)));
typedef float v8f __attribute__((ext_vector_type(8)));

#define NB 10  // N_BASIS

// out[p][n] = sum_k basis[p][k] * W[n][k] + bias[n]
//
// One wave32 handles 16 points x 128 channels via 8 WMMA 16x16 tiles,
// 3 K-chunks (K=10 padded to 12): V_WMMA_F32_16X16X4_F32 with
//   A = W-tile   (M=16 channels x K=4)
//   B = basis    (K=4 x N=16 points)
// so D gives each lane 8 *consecutive channels* of one point -> b128 stores.
__global__ __launch_bounds__(256) void bspline_wmma_kernel(
    const float* __restrict__ x, const float* __restrict__ W,
    const float* __restrict__ bias, float* __restrict__ out, int npoints)
{
    const int lane = threadIdx.x & 31;
    const int wv   = threadIdx.x >> 5;
    const int hi   = lane >> 4;   // half-wave (K-pair / M-half selector)
    const int lp   = lane & 15;   // point-in-group / channel-row-in-tile

    // ---- Hoisted W fragments (A-matrix 16x4 f32 layout) ----
    // VGPR0 = K=0 (lanes 0-15) / K=2 (lanes 16-31); VGPR1 = K=1 / K=3.
    // Lane L supplies row M = L&15 = channel 16t + lp.
    v2f wf[8][3];
#pragma unroll
    for (int t = 0; t < 8; ++t) {
        const int n = 16 * t + lp;          // channel row of this tile
        const float* wr = W + n * NB;       // W row n (row-major [128,10])
#pragma unroll
        for (int c = 0; c < 3; ++c) {
            const int k0 = 4 * c + 2 * hi;  // K index of .x element
            float e0, e1;
            if (c < 2) {                    // k0 in {0,2,4,6}: always valid
                e0 = wr[k0];
                e1 = wr[k0 + 1];
            } else {                        // k0 in {8,10}: 10,11 are padding
                e0 = hi ? 0.0f : wr[8];
                e1 = hi ? 0.0f : wr[9];
            }
            wf[t][c].x = e0;
            wf[t][c].y = e1;
        }
    }

    // ---- Hoisted bias fragments (C-matrix seed) ----
    // D layout: lane L, reg r -> channel 16t + 8*hi + r (uniform across lanes).
    v8f bfrag[8];
#pragma unroll
    for (int t = 0; t < 8; ++t) {
        const float* bp = bias + 16 * t + 8 * hi;
#pragma unroll
        for (int r = 0; r < 8; ++r) bfrag[t][r] = bp[r];
    }

    const int waves_per_blk = blockDim.x >> 5;
    const int ngroups = npoints >> 4;                    // groups of 16 points
    const int gstride = gridDim.x * waves_per_blk;
    int group = blockIdx.x * waves_per_blk + wv;

    for (; group < ngroups; group += gstride) {
        const int pbase = group << 4;

        // Lane evaluates the basis for point (pbase + lane%16); the two
        // half-waves duplicate work, matching the B-fragment layout.
        float xx = x[pbase + lp];
        xx = fminf(fmaxf(xx, 1e-9f), 1.0f);

        // Clamped uniform cubic knots: [0,0,0, 0..1 (8 pts), 1,1,1]
        const float kn[14] = {0.0f, 0.0f, 0.0f, 0.0f,
                              1.0f/7.0f, 2.0f/7.0f, 3.0f/7.0f,
                              4.0f/7.0f, 5.0f/7.0f, 6.0f/7.0f,
                              1.0f, 1.0f, 1.0f, 1.0f};

        // Degree 0 (half-open intervals; x==1.0 -> all zero, as in reference)
        float B[13];
#pragma unroll
        for (int i = 0; i < 13; ++i)
            B[i] = (xx >= kn[i] && xx < kn[i + 1]) ? 1.0f : 0.0f;

        // De Boor recursion, fully unrolled; reciprocals constant-folded,
        // zero-denominator guards become inv==0.
#pragma unroll
        for (int d = 1; d <= 3; ++d) {
#pragma unroll
            for (int i = 0; i < 13 - d; ++i) {
                const float d1 = kn[i + d]     - kn[i];
                const float d2 = kn[i + d + 1] - kn[i + 1];
                const float i1 = (d1 != 0.0f) ? 1.0f / d1 : 0.0f;
                const float i2 = (d2 != 0.0f) ? 1.0f / d2 : 0.0f;
                B[i] = (xx - kn[i]) * i1 * B[i]
                     + (kn[i + d + 1] - xx) * i2 * B[i + 1];
            }
        }

        // Pad K 10 -> 12 (constant indices: no scratch)
        float B12[12];
#pragma unroll
        for (int k = 0; k < 12; ++k) B12[k] = (k < NB) ? B[k] : 0.0f;

        // Basis fragments (B-matrix 4x16 f32 layout):
        // VGPR0 = K=0 (lanes 0-15) / K=2 (16-31); VGPR1 = K=1 / K=3; N = lp.
        v2f bb[3];
#pragma unroll
        for (int c = 0; c < 3; ++c) {
            bb[c].x = hi ? B12[4 * c + 2] : B12[4 * c + 0];
            bb[c].y = hi ? B12[4 * c + 3] : B12[4 * c + 1];
        }

        // 8 channel tiles of 16; acc seeded with bias (C-matrix).
        // Lane L writes point p = pbase + (L&15), channels 16t+8*hi+[0..7]:
        // 8 consecutive floats -> two b128 non-temporal stores.
        float* op = out + (size_t)(pbase + lp) * 128 + 8 * hi;
#pragma unroll
        for (int t = 0; t < 8; ++t) {
            v8f acc = bfrag[t];
            acc = __builtin_amdgcn_wmma_f32_16x16x4_f32(
                false, wf[t][0], false, bb[0], (short)0, acc, false, false);
            acc = __builtin_amdgcn_wmma_f32_16x16x4_f32(
                false, wf[t][1], false, bb[1], (short)0, acc, false, false);
            acc = __builtin_amdgcn_wmma_f32_16x16x4_f32(
                false, wf[t][2], false, bb[2], (short)0, acc, false, false);

            v4f lo = {acc[0], acc[1], acc[2], acc[3]};
            v4f hi4 = {acc[4], acc[5], acc[6], acc[7]};
            __builtin_nontemporal_store(lo,  (v4f*)(op + 16 * t));
            __builtin_nontemporal_store(hi4, (v4f*)(op + 16 * t + 4));
        }
    }
}

extern "C" void kernel_launch(void* const* d_in, const int* in_sizes, int n_in,
                              void* d_out, int out_size, void* d_ws, size_t ws_size,
                              hipStream_t stream) {
    const float* x    = (const float*)d_in[0];   // [512, 2048] f32
    const float* W    = (const float*)d_in[1];   // [128, 10]   f32
    const float* bias = (const float*)d_in[2];   // [128]       f32
    float*       out  = (float*)d_out;           // [512, 2048, 128] f32

    const int npoints = in_sizes[0];             // 512*2048 = 1,048,576
    (void)n_in; (void)out_size; (void)d_ws; (void)ws_size;

    // 4096 blocks x 8 waves = 32768 waves; 65536 point-groups -> 2 groups/wave.
    // Output-write bound (~512 MB): keep many waves in flight on the store path.
    bspline_wmma_kernel<<<4096, 256, 0, stream>>>(x, W, bias, out, npoints);
}